// DehazeBlock_33706903339163
// MI455X (gfx1250) — compile-verified
//
#include <hip/hip_runtime.h>

typedef __attribute__((ext_vector_type(2))) float v2f;
typedef __attribute__((ext_vector_type(4))) float v4f;
typedef __attribute__((ext_vector_type(8))) float v8f;

#define NBATCH 16
#define CCH    64
#define HW     256
#define NSLOT  256
#define BN_EPS 1e-5f

// ---------------------------------------------------------------------------
// Workspace layout (bytes):
//   [0, 147456)                    float  Wp[9][32][64][2]  (pair-interleaved weights)
//   [147456, +131072)              double psum[64][NSLOT]
//   [278528, +131072)              double psq [64][NSLOT]
//   [409600, +256)                 float  mean[64]
//   [409856, +256)                 float  rstd[64]
// Conv output is staged in the y-half of d_out (channels 64..127), then
// overwritten in place by the sorted result in the final kernel.
// ---------------------------------------------------------------------------
#define WP_FLOATS   (9 * 64 * 64)
#define PSUM_OFF    (WP_FLOATS * 4)
#define PSQ_OFF     (PSUM_OFF + 64 * NSLOT * 8)
#define MEAN_OFF    (PSQ_OFF + 64 * NSLOT * 8)

// LDS tile: [row(3)][cpair(32)][col(132)][2 floats]  (padded, pair-interleaved)
#define TCOLS       132
#define PAIR_STRIDE (TCOLS * 8)          // 1056 B
#define ROW_STRIDE  (32 * PAIR_STRIDE)   // 33792 B
#define TILE_BYTES  (3 * ROW_STRIDE)     // 101376 B

__global__ __launch_bounds__(256) void zero_stats_kernel(double* psum, double* psq) {
    int i = blockIdx.x * 256 + threadIdx.x;   // 64*NSLOT = 16384 entries each
    psum[i] = 0.0;
    psq[i]  = 0.0;
}

// Repack OIHW weights into Wp[tap][ci/2][co][ci&1] so the conv kernel's A
// fragment is a single global_load_b64 with compile-time immediate offsets.
__global__ __launch_bounds__(256) void weight_repack_kernel(
    const float* __restrict__ w, float* __restrict__ wp)
{
    const int t   = blockIdx.x * 256 + threadIdx.x;  // 36864 total
    const int co  = t & 63;
    const int ci  = (t >> 6) & 63;
    const int tap = t >> 12;
    wp[((size_t)(tap * 32 + (ci >> 1)) * 64 + co) * 2 + (ci & 1)] =
        w[((size_t)co * 64 + ci) * 9 + tap];
}

// Dilated 3x3 conv (d=2, same padding) as 9 per-tap GEMMs via
// v_wmma_f32_16x16x4_f32. One wave = 16 co x 64 px (4 accumulators sharing
// each A fragment). 8 waves/block = 64 co x 128 px of one image row.
// Input patch (3 rows x 64 ch x 132 cols, zero-padded, pair-interleaved)
// staged once in LDS and shared by all 4 co-tile waves.
__global__ __launch_bounds__(256) void conv_wmma_kernel(
    const float* __restrict__ x, const float* __restrict__ wp,
    const float* __restrict__ bias, float* __restrict__ out,
    double* __restrict__ psum, double* __restrict__ psq)
{
    extern __shared__ char smem[];
    __shared__ float lsum[64];
    __shared__ float lsq[64];

    const int tid  = threadIdx.x;
    const int lane = tid & 31;
    const int wv   = tid >> 5;
    const int hi   = lane >> 4;   // K half: lanes 0-15 -> K0/K1, 16-31 -> K2/K3
    const int m    = lane & 15;   // A: M row (co); B: N col (pixel)

    const int bi = blockIdx.x;
    const int wg = bi & 1;            // 128-pixel group in row
    const int h  = (bi >> 1) & 255;   // row
    const int b  = bi >> 9;           // batch

    const int pbase = wg * 128;

    // ---- fill LDS tile (boundary handling lives only here) ----
    if (tid < 64) { lsum[tid] = 0.0f; lsq[tid] = 0.0f; }
    const float* xb = x + (size_t)b * CCH * HW * HW;
    #pragma unroll 1
    for (int idx = tid; idx < 3 * CCH * TCOLS; idx += 256) {  // 25344 elems, 99 iters
        const int col = idx % TCOLS;
        const int t2  = idx / TCOLS;
        const int ci  = t2 & 63;
        const int row = t2 >> 6;
        const int hp  = h + row * 2 - 2;
        const int wq  = pbase + col - 2;
        float v = 0.0f;
        if ((unsigned)hp < (unsigned)HW && (unsigned)wq < (unsigned)HW)
            v = xb[((size_t)ci * HW + hp) * HW + wq];
        *(float*)(smem + ((size_t)(row * 32 + (ci >> 1)) * TCOLS + col) * 8 +
                  (ci & 1) * 4) = v;
    }
    __syncthreads();

    const int c0 = (wv & 3) * 16;   // co tile base
    const int pg = wv >> 2;         // which 64-px group

    // Per-fragment LDS base (biased by -16 so all unrolled offsets are >= 0)
    int bbm[4];
    #pragma unroll
    for (int f = 0; f < 4; ++f)
        bbm[f] = ((pg * 64 + f * 16 + m + 2) << 3) + hi * PAIR_STRIDE - 16;

    // A base: float index 2*(c0+m) + 128*hi; per-(tap,chunk) offsets constant
    const float* wpA = wp + (size_t)(c0 + m) * 2 + hi * 128;

    v8f acc[4] = {};

    #pragma unroll
    for (int tap = 0; tap < 9; ++tap) {
        const int row  = tap / 3;
        const int toff = row * ROW_STRIDE + (tap % 3) * 16;  // includes +16 bias
        const float* wt = wpA + tap * 4096;
        #pragma unroll
        for (int kb = 0; kb < CCH; kb += 4) {
            const int cp = kb >> 1;
            const v2f af = *(const v2f*)(wt + cp * 128);
            #pragma unroll
            for (int f = 0; f < 4; ++f) {
                const v2f bf = *(const v2f*)(smem + bbm[f] + toff + cp * PAIR_STRIDE);
                acc[f] = __builtin_amdgcn_wmma_f32_16x16x4_f32(
                    false, af, false, bf, (short)0, acc[f], false, false);
            }
        }
    }

    // C/D layout: VGPR r -> lanes 0-15: (M=r, N=lane); lanes 16-31: (M=r+8, N-16)
    #pragma unroll
    for (int r = 0; r < 8; ++r) {
        const int c  = c0 + r + hi * 8;
        const float bc = bias[c];
        float s = 0.0f, q = 0.0f;
        #pragma unroll
        for (int f = 0; f < 4; ++f) {
            const float v = acc[f][r] + bc;
            const int px = pbase + pg * 64 + f * 16 + m;
            // stage conv output in the y half of d_out (channel offset +64)
            out[(((size_t)b * 128 + 64 + c) * HW + h) * HW + px] = v;
            s += v;
            q += v * v;
        }
        // channel is uniform across each 16-lane half -> shuffle-reduce pixels
        #pragma unroll
        for (int mask = 1; mask <= 8; mask <<= 1) {
            s += __shfl_xor(s, mask, 32);
            q += __shfl_xor(q, mask, 32);
        }
        if (m == 0) {
            atomicAdd(&lsum[c], s);
            atomicAdd(&lsq[c],  q);
        }
    }
    __syncthreads();

    if (tid < 64) {
        const int slot = blockIdx.x & (NSLOT - 1);
        atomicAdd(&psum[tid * NSLOT + slot], (double)lsum[tid]);
        atomicAdd(&psq [tid * NSLOT + slot], (double)lsq[tid]);
    }
}

__global__ void stats_reduce_kernel(const double* __restrict__ psum,
                                    const double* __restrict__ psq,
                                    float* __restrict__ mean,
                                    float* __restrict__ rstd)
{
    const int c = threadIdx.x;
    if (c >= 64) return;
    double s = 0.0, q = 0.0;
    for (int i = 0; i < NSLOT; ++i) {
        s += psum[c * NSLOT + i];
        q += psq [c * NSLOT + i];
    }
    const double n  = (double)NBATCH * HW * HW;
    const double mu = s / n;
    const double va = q / n - mu * mu;
    mean[c] = (float)mu;
    rstd[c] = rsqrtf((float)va + BN_EPS);
}

// BN(+gamma,beta) + ReLU + 4x4 block sort. One thread owns one 4x4 block:
// reads 16 conv values from the y region, writes `a`, writes sorted `y`
// back in place over the same 16 addresses.
__global__ __launch_bounds__(256) void bn_relu_sort_kernel(
    float* __restrict__ out,
    const float* __restrict__ mean, const float* __restrict__ rstd,
    const float* __restrict__ gamma, const float* __restrict__ beta)
{
    const int t  = blockIdx.x * 256 + threadIdx.x;
    const int wb = t & 63;
    const int hb = (t >> 6) & 63;
    const int c  = (t >> 12) & 63;
    const int b  = t >> 18;

    const float mu = mean[c];
    const float rs = rstd[c];
    const float g  = gamma[c];
    const float be = beta[c];

    const size_t ybase = (((size_t)b * 128 + 64 + c) * HW + hb * 4) * HW + wb * 4;
    const size_t abase = (((size_t)b * 128 + c)      * HW + hb * 4) * HW + wb * 4;

    float v[16];
    #pragma unroll
    for (int r = 0; r < 4; ++r) {
        v4f q = *(const v4f*)(out + ybase + (size_t)r * HW);
        v[r * 4 + 0] = q.x; v[r * 4 + 1] = q.y;
        v[r * 4 + 2] = q.z; v[r * 4 + 3] = q.w;
    }

    #pragma unroll
    for (int i = 0; i < 16; ++i) {
        const float z = (v[i] - mu) * rs;
        v[i] = fmaxf(g * z + be, 0.0f);
    }

    #pragma unroll
    for (int r = 0; r < 4; ++r) {
        v4f q;
        q.x = v[r * 4 + 0]; q.y = v[r * 4 + 1];
        q.z = v[r * 4 + 2]; q.w = v[r * 4 + 3];
        *(v4f*)(out + abase + (size_t)r * HW) = q;
    }

    // 16-element bitonic sort, ascending, fully unrolled in VGPRs
    #pragma unroll
    for (int k = 2; k <= 16; k <<= 1) {
        #pragma unroll
        for (int j = k >> 1; j > 0; j >>= 1) {
            #pragma unroll
            for (int i = 0; i < 16; ++i) {
                const int l = i ^ j;
                if (l > i) {
                    const bool up = ((i & k) == 0);
                    const float a0 = v[i], a1 = v[l];
                    const bool sw = up ? (a0 > a1) : (a0 < a1);
                    v[i] = sw ? a1 : a0;
                    v[l] = sw ? a0 : a1;
                }
            }
        }
    }

    #pragma unroll
    for (int r = 0; r < 4; ++r) {
        v4f q;
        q.x = v[r * 4 + 0]; q.y = v[r * 4 + 1];
        q.z = v[r * 4 + 2]; q.w = v[r * 4 + 3];
        *(v4f*)(out + ybase + (size_t)r * HW) = q;
    }
}

extern "C" void kernel_launch(void* const* d_in, const int* in_sizes, int n_in,
                              void* d_out, int out_size, void* d_ws, size_t ws_size,
                              hipStream_t stream) {
    const float* x      = (const float*)d_in[0];
    const float* conv_w = (const float*)d_in[1];
    const float* conv_b = (const float*)d_in[2];
    const float* gamma  = (const float*)d_in[3];
    const float* beta   = (const float*)d_in[4];
    float* out = (float*)d_out;

    char* ws = (char*)d_ws;
    float*  wpk  = (float*) ws;
    double* psum = (double*)(ws + PSUM_OFF);
    double* psq  = (double*)(ws + PSQ_OFF);
    float*  mean = (float*) (ws + MEAN_OFF);
    float*  rstd = mean + 64;

    // 1) zero stats partials (2 * 64 * 256 doubles)
    zero_stats_kernel<<<64, 256, 0, stream>>>(psum, psq);

    // 2) repack weights into pair-interleaved WMMA-friendly layout
    weight_repack_kernel<<<WP_FLOATS / 256, 256, 0, stream>>>(conv_w, wpk);

    // 3) conv via WMMA: 16 batches * 256 rows * 2 groups-of-128-px = 8192 blocks
    conv_wmma_kernel<<<NBATCH * HW * 2, 256, TILE_BYTES, stream>>>(
        x, wpk, conv_b, out, psum, psq);

    // 4) finalize per-channel mean / rstd
    stats_reduce_kernel<<<1, 64, 0, stream>>>(psum, psq, mean, rstd);

    // 5) BN + ReLU + 4x4 block sort: 16*64*64*64 = 4,194,304 blocks-of-16
    bn_relu_sort_kernel<<<(NBATCH * CCH * 64 * 64) / 256, 256, 0, stream>>>(
        out, mean, rstd, gamma, beta);
}